// LongformerAttention_23373212025415
// MI455X (gfx1250) — compile-verified
//
#include <hip/hip_runtime.h>
#include <math.h>

typedef __attribute__((ext_vector_type(2))) float v2f;
typedef __attribute__((ext_vector_type(8))) float v8f;

__device__ __forceinline__ v8f wmma_f32(v2f a, v2f b, v8f c) {
  // 8 args: (neg_a, A, neg_b, B, c_mod, C, reuse_a, reuse_b)
  return __builtin_amdgcn_wmma_f32_16x16x4_f32(false, a, false, b, (short)0, c,
                                               false, false);
}

// =====================================================================
// GEMM:  C[M,N] = X[M,K] @ W[N,K]^T + bias[N]     (all fp32)
// Block tile 128x64, 8 waves (4x2), each wave 32x32 (2x2 WMMA tiles).
// Register double-buffered K panels + L2 prefetch of panel k0+2*BK.
// =====================================================================
#define GM_BM 128
#define GM_BN 64
#define GM_BK 32
#define GM_PITCH 36  // even -> 8B-aligned v2f frags; 36%4==0 -> float4 stores

__global__ __launch_bounds__(256) void gemm_xwt_bias(
    const float* __restrict__ X, const float* __restrict__ W,
    const float* __restrict__ bias, float* __restrict__ C,
    int M, int N, int K) {
  __shared__ float Xs[GM_BM * GM_PITCH];
  __shared__ float Ws[GM_BN * GM_PITCH];

  const int tid = threadIdx.x;
  const int lane = tid & 31;
  const int wave = tid >> 5;   // 0..7
  const int wm = wave >> 1;    // 0..3 : 32-row strip
  const int wn = wave & 1;     // 0..1 : 32-col strip
  const int half = lane >> 4;  // A/B frag K-half select
  const int l16 = lane & 15;

  const int m0 = blockIdx.y * GM_BM;
  const int n0 = blockIdx.x * GM_BN;

  v8f acc[2][2];
#pragma unroll
  for (int i = 0; i < 2; ++i)
#pragma unroll
    for (int j = 0; j < 2; ++j) acc[i][j] = {0.f, 0.f, 0.f, 0.f, 0.f, 0.f, 0.f, 0.f};

  // Per-thread staging coordinates (float4 chunks).
  float4 xr[4], wr[2];
  auto load_panel = [&](int k0) {
#pragma unroll
    for (int i = 0; i < 4; ++i) {
      int chunk = tid + i * 256;       // 0..1023 -> 128x32 X tile
      int r = chunk >> 3;
      int c = (chunk & 7) << 2;
      xr[i] = *(const float4*)(X + (size_t)(m0 + r) * K + k0 + c);
    }
#pragma unroll
    for (int i = 0; i < 2; ++i) {
      int chunk = tid + i * 256;       // 0..511 -> 64x32 W tile
      int r = chunk >> 3;
      int c = (chunk & 7) << 2;
      wr[i] = *(const float4*)(W + (size_t)(n0 + r) * K + k0 + c);
    }
  };

  load_panel(0);
  for (int k0 = 0; k0 < K; k0 += GM_BK) {
    __syncthreads();  // previous panel's LDS reads complete
#pragma unroll
    for (int i = 0; i < 4; ++i) {
      int chunk = tid + i * 256;
      int r = chunk >> 3;
      int c = (chunk & 7) << 2;
      *(float4*)&Xs[r * GM_PITCH + c] = xr[i];
    }
#pragma unroll
    for (int i = 0; i < 2; ++i) {
      int chunk = tid + i * 256;
      int r = chunk >> 3;
      int c = (chunk & 7) << 2;
      *(float4*)&Ws[r * GM_PITCH + c] = wr[i];
    }
    // Issue next panel's global loads now; they retire during the WMMA phase.
    if (k0 + GM_BK < K) load_panel(k0 + GM_BK);
    // L2 prefetch of the panel after next (gfx1250 global_prefetch_b8).
    if (k0 + 2 * GM_BK < K) {
      int r = tid >> 3;
      int c = (tid & 7) << 2;
      __builtin_prefetch(X + (size_t)(m0 + r) * K + k0 + 2 * GM_BK + c, 0, 1);
      __builtin_prefetch(W + (size_t)(n0 + (r >> 1)) * K + k0 + 2 * GM_BK + c, 0, 1);
    }
    __syncthreads();  // staged panel visible

#pragma unroll
    for (int kk = 0; kk < GM_BK; kk += 4) {
      const int koff = kk + 2 * half;
      v2f a0 = *(const v2f*)&Xs[(wm * 32 + l16) * GM_PITCH + koff];
      v2f a1 = *(const v2f*)&Xs[(wm * 32 + 16 + l16) * GM_PITCH + koff];
      v2f b0 = *(const v2f*)&Ws[(wn * 32 + l16) * GM_PITCH + koff];
      v2f b1 = *(const v2f*)&Ws[(wn * 32 + 16 + l16) * GM_PITCH + koff];
      acc[0][0] = wmma_f32(a0, b0, acc[0][0]);
      acc[0][1] = wmma_f32(a0, b1, acc[0][1]);
      acc[1][0] = wmma_f32(a1, b0, acc[1][0]);
      acc[1][1] = wmma_f32(a1, b1, acc[1][1]);
    }
  }

  // Epilogue: bias + store. C layout: VGPR j -> row j + 8*half, col = l16.
#pragma unroll
  for (int mi = 0; mi < 2; ++mi)
#pragma unroll
    for (int ni = 0; ni < 2; ++ni) {
      const int col = n0 + wn * 32 + ni * 16 + l16;
      const float bv = bias[col];
#pragma unroll
      for (int j = 0; j < 8; ++j) {
        const int row = m0 + wm * 32 + mi * 16 + j + 8 * half;
        C[(size_t)row * N + col] = acc[mi][ni][j] + bv;
      }
    }
}

// =====================================================================
// Longformer attention, flash-style online softmax with f32 WMMA.
// One block = one (b,h) pair x 128-query tile. 8 waves x 16 queries.
// q,k,v,ctx stored as [B, S, H*DH] so head h = column slice [64h, 64h+64).
// =====================================================================
#define QT 128
#define NGLOB 204   // max(1, int(2048*0.1))
#define WHALF 64    // window_size // 2
#define ATT_SCALE 0.125f
#define KV_PITCH 68  // even, %4==0
#define VT_PITCH 18  // even
#define P_PITCH 18   // even

__global__ __launch_bounds__(256) void lf_attention(
    const float* __restrict__ Q, const float* __restrict__ Km,
    const float* __restrict__ Vm, float* __restrict__ Ctx,
    int B, int S, int Dm) {
  __shared__ float Ks[16 * KV_PITCH];       // K tile  [key][d]
  __shared__ float VsT[64 * VT_PITCH];      // V tile transposed [d][key]
  __shared__ float Ps[8][16 * P_PITCH];     // per-wave P tile [q][key]

  const int tid = threadIdx.x;
  const int lane = tid & 31;
  const int wave = tid >> 5;
  const int half = lane >> 4;
  const int l16 = lane & 15;

  const int q0 = blockIdx.x * QT;
  const int bh = blockIdx.y;  // b*16 + h
  const int b = bh >> 4;
  const int ho = bh & 15;

  // Preload Q fragments: A-matrix 16x64, 16 k-steps of 4.
  const int qrow = q0 + wave * 16 + l16;
  const float* qbase = Q + (size_t)(b * S + qrow) * Dm + ho * 64;
  v2f qa[16];
#pragma unroll
  for (int t = 0; t < 16; ++t) qa[t] = *(const v2f*)(qbase + 4 * t + 2 * half);

  v8f o[4];
#pragma unroll
  for (int t = 0; t < 4; ++t) o[t] = {0.f, 0.f, 0.f, 0.f, 0.f, 0.f, 0.f, 0.f};
  float mrow[8], lrow[8];
#pragma unroll
  for (int j = 0; j < 8; ++j) { mrow[j] = -1e30f; lrow[j] = 0.f; }

  auto process_tile = [&](int kt) {
    const int k0 = kt * 16;
    __syncthreads();
    {  // cooperative staging: 256 threads, one float4 each per tile
      const int r = tid >> 4;          // 0..15 key row
      const int c = (tid & 15) << 2;   // 0..60 d col
      float4 kv = *(const float4*)(Km + (size_t)(b * S + k0 + r) * Dm + ho * 64 + c);
      *(float4*)&Ks[r * KV_PITCH + c] = kv;
      float4 vv = *(const float4*)(Vm + (size_t)(b * S + k0 + r) * Dm + ho * 64 + c);
      VsT[(c + 0) * VT_PITCH + r] = vv.x;
      VsT[(c + 1) * VT_PITCH + r] = vv.y;
      VsT[(c + 2) * VT_PITCH + r] = vv.z;
      VsT[(c + 3) * VT_PITCH + r] = vv.w;
      // L2 prefetch of the (likely) next key tile.
      const int kn = (kt + 1 < (S >> 4)) ? (kt + 1) * 16 : k0;
      __builtin_prefetch(Km + (size_t)(b * S + kn + r) * Dm + ho * 64 + c, 0, 1);
      __builtin_prefetch(Vm + (size_t)(b * S + kn + r) * Dm + ho * 64 + c, 0, 1);
    }
    __syncthreads();

    // S_tile[16q x 16k] = Q(16x64) @ K^T(64x16)
    v8f sc = {0.f, 0.f, 0.f, 0.f, 0.f, 0.f, 0.f, 0.f};
#pragma unroll
    for (int t = 0; t < 16; ++t) {
      v2f kb = *(const v2f*)&Ks[l16 * KV_PITCH + 4 * t + 2 * half];
      sc = wmma_f32(qa[t], kb, sc);
    }

    // mask + online softmax (rows live across the 16-lane half)
    const int jk = k0 + l16;
#pragma unroll
    for (int j = 0; j < 8; ++j) {
      const int i = q0 + wave * 16 + j + 8 * half;
      int di = i - jk; di = di < 0 ? -di : di;
      const bool att = (di <= WHALF) || (i < NGLOB) || (jk < NGLOB);
      float s = sc[j] * ATT_SCALE + (att ? 0.f : -10000.f);
      float rm = s;
      rm = fmaxf(rm, __shfl_xor(rm, 1));
      rm = fmaxf(rm, __shfl_xor(rm, 2));
      rm = fmaxf(rm, __shfl_xor(rm, 4));
      rm = fmaxf(rm, __shfl_xor(rm, 8));
      const float mnew = fmaxf(mrow[j], rm);
      const float p = __expf(s - mnew);
      float rs = p;
      rs += __shfl_xor(rs, 1);
      rs += __shfl_xor(rs, 2);
      rs += __shfl_xor(rs, 4);
      rs += __shfl_xor(rs, 8);
      const float alpha = __expf(mrow[j] - mnew);
      lrow[j] = lrow[j] * alpha + rs;
      mrow[j] = mnew;
#pragma unroll
      for (int t = 0; t < 4; ++t) o[t][j] *= alpha;
      Ps[wave][(j + 8 * half) * P_PITCH + l16] = p;  // C-layout -> LDS
    }
    __builtin_amdgcn_wave_barrier();  // DS in-order within wave; pin ordering

    // O(16x64) += P(16x16) @ V(16x64)
#pragma unroll
    for (int kk = 0; kk < 16; kk += 4) {
      v2f pa = *(const v2f*)&Ps[wave][l16 * P_PITCH + kk + 2 * half];
#pragma unroll
      for (int t = 0; t < 4; ++t) {
        v2f vb = *(const v2f*)&VsT[(t * 16 + l16) * VT_PITCH + kk + 2 * half];
        o[t] = wmma_f32(pa, vb, o[t]);
      }
    }
  };

  // Key-tile schedule (block-uniform): global tiles + sliding window tiles.
  int tEnd1, tBeg2, tEnd2;
  if (q0 < NGLOB) {  // block contains global queries -> attend everything
    tEnd1 = S >> 4; tBeg2 = 0; tEnd2 = 0;
  } else {
    tEnd1 = (NGLOB + 15) >> 4;                       // 13 tiles: keys [0,208)
    tBeg2 = (q0 - WHALF) >> 4;
    if (tBeg2 < tEnd1) tBeg2 = tEnd1;
    tEnd2 = ((q0 + QT - 1 + WHALF) >> 4) + 1;
    if (tEnd2 > (S >> 4)) tEnd2 = S >> 4;
  }
  for (int kt = 0; kt < tEnd1; ++kt) process_tile(kt);
  for (int kt = tBeg2; kt < tEnd2; ++kt) process_tile(kt);

  // Normalize and write ctx (coalesced per half-wave)
#pragma unroll
  for (int j = 0; j < 8; ++j) {
    const int row = q0 + wave * 16 + j + 8 * half;
    const float inv = 1.f / lrow[j];
#pragma unroll
    for (int t = 0; t < 4; ++t)
      Ctx[(size_t)(b * S + row) * Dm + ho * 64 + t * 16 + l16] = o[t][j] * inv;
  }
}

// =====================================================================
extern "C" void kernel_launch(void* const* d_in, const int* in_sizes, int n_in,
                              void* d_out, int out_size, void* d_ws, size_t ws_size,
                              hipStream_t stream) {
  const float* hs = (const float*)d_in[0];
  const float* Wq = (const float*)d_in[1];
  const float* bq = (const float*)d_in[2];
  const float* Wk = (const float*)d_in[3];
  const float* bk = (const float*)d_in[4];
  const float* Wv = (const float*)d_in[5];
  const float* bv = (const float*)d_in[6];
  const float* Wo = (const float*)d_in[7];
  const float* bo = (const float*)d_in[8];

  const int B = 2, S = 2048, Dm = 1024;
  const int M = B * S;
  const size_t MN = (size_t)M * Dm;

  float* ws = (float*)d_ws;
  float* q = ws + 0 * MN;
  float* k = ws + 1 * MN;
  float* v = ws + 2 * MN;
  float* ctx = ws + 3 * MN;

  dim3 gg(Dm / GM_BN, M / GM_BM);  // (16, 32)
  gemm_xwt_bias<<<gg, 256, 0, stream>>>(hs, Wq, bq, q, M, Dm, Dm);
  gemm_xwt_bias<<<gg, 256, 0, stream>>>(hs, Wk, bk, k, M, Dm, Dm);
  gemm_xwt_bias<<<gg, 256, 0, stream>>>(hs, Wv, bv, v, M, Dm, Dm);

  dim3 ga(S / QT, B * 16);  // (16, 32)
  lf_attention<<<ga, 256, 0, stream>>>(q, k, v, ctx, B, S, Dm);

  gemm_xwt_bias<<<gg, 256, 0, stream>>>(ctx, Wo, bo, (float*)d_out, M, Dm, Dm);
}